// LacationSensitiveAttention_51582557225634
// MI455X (gfx1250) — compile-verified
//
#include <hip/hip_runtime.h>
#include <hip/hip_bf16.h>
#include <math.h>

// Problem constants (from reference)
#define B_ 64
#define T_ 2048
#define E_ 512
#define D_ 1024
#define U_ 128
#define TTILES_ (T_ / 16)   // 128

typedef float v2f __attribute__((ext_vector_type(2)));
typedef float v8f __attribute__((ext_vector_type(8)));

// ---------------------------------------------------------------------------
// Kernel 0: q_proj[b][u] = sum_d queries[b,0,d] * Wq[u,d]
// ---------------------------------------------------------------------------
__global__ __launch_bounds__(128) void k_qproj(const float* __restrict__ q,
                                               const float* __restrict__ Wq,
                                               float* __restrict__ qproj) {
    const int b = blockIdx.x;
    const int u = threadIdx.x;
    const float4* qr = reinterpret_cast<const float4*>(q + (size_t)b * D_);
    const float4* wr = reinterpret_cast<const float4*>(Wq + (size_t)u * D_);
    float acc = 0.f;
    for (int d = 0; d < D_ / 4; ++d) {
        float4 a = qr[d], w = wr[d];
        acc += a.x * w.x + a.y * w.y + a.z * w.z + a.w * w.w;
    }
    qproj[b * U_ + u] = acc;
}

// ---------------------------------------------------------------------------
// Kernel 1: fused proj_key GEMM (fp32 WMMA) + conv1d + bias + tanh + v-dot
//           -> masked score[b][t] into workspace.
// Block = 256 threads (8 waves). Block handles (b, 16-row t-tile).
// Wave w handles u-tile [16w, 16w+16). K = E = 512, stepped by 4 (WMMA K).
// ---------------------------------------------------------------------------
__global__ __launch_bounds__(256) void k_scores(
    const float* __restrict__ enc,     // (B,T,E)
    const float* __restrict__ Wk,      // (U,E)
    const float* __restrict__ qproj,   // (B,U) workspace
    const float* __restrict__ prev,    // (B,T)
    const float* __restrict__ conv_w,  // (U,1,3)
    const float* __restrict__ conv_b,  // (U)
    const float* __restrict__ biasb,   // (U)
    const float* __restrict__ v_w,     // (1,U)
    const float* __restrict__ v_b,     // (1)
    const int*   __restrict__ lengths, // (B)
    float* __restrict__ score)         // (B,T) workspace
{
    __shared__ float As[16 * E_];      // 32 KB: enc tile, 16 t-rows x 512 e
    __shared__ float sc[16];

    const int b  = blockIdx.x / TTILES_;
    const int t0 = (blockIdx.x % TTILES_) * 16;
    const int tid = threadIdx.x;

    // Stage the A tile (16 x 512 fp32) into LDS with float4 loads.
    {
        const float4* enc4 = reinterpret_cast<const float4*>(
            enc + ((size_t)b * T_ + t0) * E_);
        float4* As4 = reinterpret_cast<float4*>(As);
        for (int i = tid; i < 16 * (E_ / 4); i += 256)
            As4[i] = enc4[i];
    }
    if (tid < 16) sc[tid] = 0.f;
    __syncthreads();

    const int wave = tid >> 5;
    const int lane = tid & 31;
    const int n    = lane & 15;   // column of the 16x16 tile (u offset)
    const int half = lane >> 4;   // 0: K 0..1 / M 0..7 ; 1: K 2..3 / M 8..15
    const int u0   = wave * 16;
    const int u    = u0 + n;
    const int koff = half * 2;

    // A-frag (16x4 fp32): lane (&15) = M row; VGPR0/1 = K (koff, koff+1)
    // B-frag (4x16 fp32): lane (&15) = N col; VGPR0/1 = K (koff, koff+1)
    //   B[k][n] = Wk[u0+n][k]  (Wk row-major over E -> contiguous float2)
    v8f c = {};
    const float* wkrow = Wk + (size_t)u * E_;
    const float* arow  = As + (size_t)n * E_;
#pragma unroll 4
    for (int k = 0; k < E_; k += 4) {
        v2f a, bf;
        a.x  = arow[k + koff];
        a.y  = arow[k + koff + 1];
        bf.x = wkrow[k + koff];
        bf.y = wkrow[k + koff + 1];
        c = __builtin_amdgcn_wmma_f32_16x16x4_f32(
                /*neg_a=*/false, a, /*neg_b=*/false, bf,
                /*c_mod=*/(short)0, c, /*reuse_a=*/false, /*reuse_b=*/false);
    }

    // Fused epilogue: + q_proj + bias + conv1d(prev) -> tanh -> * v_w[u]
    const float qp  = qproj[b * U_ + u];
    const float add = qp + biasb[u] + conv_b[u];
    const float cw0 = conv_w[u * 3 + 0];
    const float cw1 = conv_w[u * 3 + 1];
    const float cw2 = conv_w[u * 3 + 2];
    const float vw  = v_w[u];
    const float* prow = prev + (size_t)b * T_;

#pragma unroll
    for (int r = 0; r < 8; ++r) {
        const int m = r + 8 * half;          // C layout: VGPR r -> M = r | r+8
        const int t = t0 + m;
        const float pm1 = (t > 0)      ? prow[t - 1] : 0.f;
        const float p0  =                prow[t];
        const float pp1 = (t + 1 < T_) ? prow[t + 1] : 0.f;
        const float val = c[r] + add + cw0 * pm1 + cw1 * p0 + cw2 * pp1;
        float contrib = tanhf(val) * vw;
        // Sum over the 16 lanes of this half (the 16 u's of the wave's tile)
        contrib += __shfl_xor(contrib, 1, 32);
        contrib += __shfl_xor(contrib, 2, 32);
        contrib += __shfl_xor(contrib, 4, 32);
        contrib += __shfl_xor(contrib, 8, 32);
        if (n == 0) atomicAdd(&sc[m], contrib);   // ds_add across 8 waves
    }
    __syncthreads();

    if (tid < 16) {
        const int t = t0 + tid;
        float s = sc[tid] + v_b[0];
        if (t >= lengths[b]) s = -INFINITY;
        score[(size_t)b * T_ + t] = s;
    }
}

// ---------------------------------------------------------------------------
// Kernel 2: masked softmax over T per batch row -> attn (into d_out)
// ---------------------------------------------------------------------------
__global__ __launch_bounds__(256) void k_softmax(const float* __restrict__ score,
                                                 float* __restrict__ attn) {
    const int b = blockIdx.x;
    const int tid = threadIdx.x;
    __shared__ float red[8];
    __shared__ float bmax, bsum;
    const float* srow = score + (size_t)b * T_;

    float m = -INFINITY;
    for (int t = tid; t < T_; t += 256) m = fmaxf(m, srow[t]);
    for (int off = 1; off < 32; off <<= 1) m = fmaxf(m, __shfl_xor(m, off, 32));
    if ((tid & 31) == 0) red[tid >> 5] = m;
    __syncthreads();
    if (tid == 0) {
        float v = red[0];
        for (int i = 1; i < 8; ++i) v = fmaxf(v, red[i]);
        bmax = v;
    }
    __syncthreads();
    const float mx = bmax;

    float s = 0.f;
    for (int t = tid; t < T_; t += 256) s += expf(srow[t] - mx);
    for (int off = 1; off < 32; off <<= 1) s += __shfl_xor(s, off, 32);
    if ((tid & 31) == 0) red[tid >> 5] = s;
    __syncthreads();
    if (tid == 0) {
        float v = 0.f;
        for (int i = 1; i < 8; ++i) v += red[i];
        bsum = v + red[0];
    }
    __syncthreads();
    const float inv = 1.f / bsum;

    for (int t = tid; t < T_; t += 256)
        attn[(size_t)b * T_ + t] = expf(srow[t] - mx) * inv;
}

// ---------------------------------------------------------------------------
// Kernel 3a: partial context over a 256-row T chunk (streaming, fp32 FMA)
// grid (B, 8), block 128 threads, one float4 of E per thread.
// ---------------------------------------------------------------------------
__global__ __launch_bounds__(128) void k_ctx_part(const float* __restrict__ enc,
                                                  const float* __restrict__ attn,
                                                  float* __restrict__ part) {
    const int b = blockIdx.x;
    const int chunk = blockIdx.y;           // 8 chunks of 256 t's
    const int tid = threadIdx.x;            // float4 index over E (512/4=128)
    __shared__ float aS[256];

    const float* arow = attn + (size_t)b * T_ + chunk * 256;
    for (int i = tid; i < 256; i += 128) aS[i] = arow[i];
    __syncthreads();

    const float4* enc4 = reinterpret_cast<const float4*>(
        enc + ((size_t)b * T_ + (size_t)chunk * 256) * E_);
    float4 acc = {0.f, 0.f, 0.f, 0.f};
    for (int t = 0; t < 256; ++t) {
        const float a = aS[t];
        const float4 ev = enc4[(size_t)t * (E_ / 4) + tid];
        acc.x += a * ev.x; acc.y += a * ev.y;
        acc.z += a * ev.z; acc.w += a * ev.w;
    }
    float4* p4 = reinterpret_cast<float4*>(part + ((size_t)(b * 8 + chunk)) * E_);
    p4[tid] = acc;
}

// Kernel 3b: reduce 8 partials -> context (into d_out)
__global__ __launch_bounds__(256) void k_ctx_reduce(const float* __restrict__ part,
                                                    float* __restrict__ ctx) {
    const int idx = blockIdx.x * 256 + threadIdx.x;   // B*E = 32768
    if (idx >= B_ * E_) return;
    const int b = idx / E_;
    const int e = idx % E_;
    float s = 0.f;
    for (int c = 0; c < 8; ++c) s += part[((size_t)(b * 8 + c)) * E_ + e];
    ctx[idx] = s;
}

// ---------------------------------------------------------------------------
extern "C" void kernel_launch(void* const* d_in, const int* in_sizes, int n_in,
                              void* d_out, int out_size, void* d_ws, size_t ws_size,
                              hipStream_t stream) {
    const float* queries = (const float*)d_in[0];   // (B,1,D)
    const float* prev    = (const float*)d_in[1];   // (B,T)
    const float* enc     = (const float*)d_in[2];   // (B,T,E)
    const float* conv_w  = (const float*)d_in[3];   // (U,1,3)
    const float* conv_b  = (const float*)d_in[4];   // (U)
    const float* Wq      = (const float*)d_in[5];   // (U,D)
    const float* Wk      = (const float*)d_in[6];   // (U,E)
    const float* v_w     = (const float*)d_in[7];   // (1,U)
    const float* v_b     = (const float*)d_in[8];   // (1)
    const float* biasb   = (const float*)d_in[9];   // (U)
    const int*   lengths = (const int*)d_in[10];    // (B)

    float* ctx_out  = (float*)d_out;                // (B,1,E) = 32768 floats
    float* attn_out = (float*)d_out + (size_t)B_ * E_;  // (B,T) = 131072 floats

    // Workspace layout (floats): qproj[B*U] | score[B*T] | ctx partials[B*8*E]
    float* ws_qproj = (float*)d_ws;
    float* ws_score = ws_qproj + (size_t)B_ * U_;
    float* ws_part  = ws_score + (size_t)B_ * T_;

    k_qproj<<<B_, 128, 0, stream>>>(queries, Wq, ws_qproj);

    k_scores<<<B_ * TTILES_, 256, 0, stream>>>(
        enc, Wk, ws_qproj, prev, conv_w, conv_b, biasb, v_w, v_b, lengths,
        ws_score);

    k_softmax<<<B_, 256, 0, stream>>>(ws_score, attn_out);

    k_ctx_part<<<dim3(B_, 8), 128, 0, stream>>>(enc, attn_out, ws_part);

    k_ctx_reduce<<<(B_ * E_ + 255) / 256, 256, 0, stream>>>(ws_part, ctx_out);
}